// VBNetworkCentrality_73126113181907
// MI455X (gfx1250) — compile-verified
//
#include <hip/hip_runtime.h>
#include <hip/hip_bf16.h>

// VBNetworkCentrality on gfx1250 (MI455X).
//
// loglik = sum_edges mean_samples log(sigmoid(c[s,i]*c[s,j]) + 1e-8)
//   c[s,n] = mu[n] + exp(log_sigma[n]) * eps[s,n]
//
// Memory-bound gather/reduce: no GEMM structure -> WMMA not applicable.
// Optimization targets: node-major c layout (8 floats = 32B contiguous per
// node -> two b128 gathers per endpoint, table L2-resident at 3.2MB of 192MB),
// non-temporal b128 edge streaming (25.6MB @ 23.3TB/s ~= 1.1us floor),
// global_prefetch_b8 ahead of the stream, wave32 shuffle reductions,
// deterministic two-pass final reduction (no float atomics).

#define N_SAMPLES 8
#define TPB 256
#define MAX_EDGE_BLOCKS 1024

// Native (clang ext_vector) int4 for the non-temporal builtin.
typedef __attribute__((ext_vector_type(4))) int v4i;

// log(sigmoid(x) + 1e-8) with native v_exp_f32 / v_rcp_f32 / v_log_f32.
// x >> 0: t->0, p->1, result ~ 1e-8.  x << 0: t->inf, p->0, result = log(1e-8).
__device__ __forceinline__ float edge_ll(float x) {
    float t = __expf(-x);                       // v_exp_f32
    float p = __builtin_amdgcn_rcpf(1.0f + t);  // v_rcp_f32
    return __logf(p + 1e-8f);                   // v_log_f32
}

// Deterministic block sum (wave32). Result valid in thread 0.
__device__ __forceinline__ float block_reduce_sum(float v) {
    #pragma unroll
    for (int off = 16; off > 0; off >>= 1)
        v += __shfl_down(v, off, 32);
    __shared__ float red[TPB / 32];
    const int lane = threadIdx.x & 31;
    const int wave = threadIdx.x >> 5;
    if (lane == 0) red[wave] = v;
    __syncthreads();
    v = 0.0f;
    if (wave == 0) {
        v = (lane < (TPB / 32)) ? red[lane] : 0.0f;
        #pragma unroll
        for (int off = 4; off > 0; off >>= 1)
            v += __shfl_down(v, off, 32);
    }
    return v;
}

// Kernel 1: c[n*8 + s] = mu[n] + exp(log_sigma[n]) * eps[s*n_nodes + n]
// eps reads are coalesced across lanes for each s; output is two float4
// stores per node (node-major layout for the edge-gather phase).
__global__ __launch_bounds__(TPB) void build_c_kernel(
    const float* __restrict__ mu,
    const float* __restrict__ log_sigma,
    const float* __restrict__ eps,
    float4* __restrict__ c,        // 2 float4 per node
    int n_nodes) {
    const int n = blockIdx.x * blockDim.x + threadIdx.x;
    if (n >= n_nodes) return;
    const float m = mu[n];
    const float sg = __expf(log_sigma[n]);
    float v[N_SAMPLES];
    #pragma unroll
    for (int s = 0; s < N_SAMPLES; ++s)
        v[s] = fmaf(sg, eps[(size_t)s * (size_t)n_nodes + n], m);
    float4 lo = make_float4(v[0], v[1], v[2], v[3]);
    float4 hi = make_float4(v[4], v[5], v[6], v[7]);
    c[2 * n + 0] = lo;
    c[2 * n + 1] = hi;
}

// Kernel 2: grid-stride over edge pairs (2 int32 edges per 128-bit NT load),
// gather both endpoints' 8-sample vectors from the L2-resident table,
// accumulate log-likelihood, deterministic block partial sum.
__global__ __launch_bounds__(TPB) void edge_reduce_kernel(
    const v4i* __restrict__ epairs,    // 2 edges per element: (i0,j0,i1,j1)
    const int2* __restrict__ e2,       // single-edge view for odd tail
    const float4* __restrict__ c,      // 2 float4 per node
    float* __restrict__ partials,
    int n_pairs, int n_edges) {
    float acc = 0.0f;
    const int stride = gridDim.x * blockDim.x;
    for (int k = blockIdx.x * blockDim.x + threadIdx.x; k < n_pairs; k += stride) {
        // Stay ahead of the stream (gfx1250 global_prefetch_b8).
        if (k + 2 * stride < n_pairs)
            __builtin_prefetch(&epairs[k + 2 * stride], 0, 0);
        // Non-temporal 128-bit edge load: don't displace the c table.
        const v4i e = __builtin_nontemporal_load(&epairs[k]);

        const float4 a0 = c[2 * e.x + 0], a1 = c[2 * e.x + 1];
        const float4 b0 = c[2 * e.y + 0], b1 = c[2 * e.y + 1];
        acc += edge_ll(a0.x * b0.x); acc += edge_ll(a0.y * b0.y);
        acc += edge_ll(a0.z * b0.z); acc += edge_ll(a0.w * b0.w);
        acc += edge_ll(a1.x * b1.x); acc += edge_ll(a1.y * b1.y);
        acc += edge_ll(a1.z * b1.z); acc += edge_ll(a1.w * b1.w);

        const float4 p0 = c[2 * e.z + 0], p1 = c[2 * e.z + 1];
        const float4 q0 = c[2 * e.w + 0], q1 = c[2 * e.w + 1];
        acc += edge_ll(p0.x * q0.x); acc += edge_ll(p0.y * q0.y);
        acc += edge_ll(p0.z * q0.z); acc += edge_ll(p0.w * q0.w);
        acc += edge_ll(p1.x * q1.x); acc += edge_ll(p1.y * q1.y);
        acc += edge_ll(p1.z * q1.z); acc += edge_ll(p1.w * q1.w);
    }
    // Odd trailing edge handled by exactly one thread (deterministic).
    if ((n_edges & 1) && blockIdx.x == 0 && threadIdx.x == 0) {
        const int2 e = e2[n_edges - 1];
        const float4 a0 = c[2 * e.x + 0], a1 = c[2 * e.x + 1];
        const float4 b0 = c[2 * e.y + 0], b1 = c[2 * e.y + 1];
        acc += edge_ll(a0.x * b0.x); acc += edge_ll(a0.y * b0.y);
        acc += edge_ll(a0.z * b0.z); acc += edge_ll(a0.w * b0.w);
        acc += edge_ll(a1.x * b1.x); acc += edge_ll(a1.y * b1.y);
        acc += edge_ll(a1.z * b1.z); acc += edge_ll(a1.w * b1.w);
    }
    const float total = block_reduce_sum(acc);
    if (threadIdx.x == 0) partials[blockIdx.x] = total;
}

// Kernel 3: single-block deterministic reduction of block partials,
// then divide by n_samples (mean over axis 0).
__global__ __launch_bounds__(TPB) void final_reduce_kernel(
    const float* __restrict__ partials, int n,
    const int* __restrict__ n_samples,
    float* __restrict__ out) {
    float acc = 0.0f;
    for (int k = threadIdx.x; k < n; k += TPB)
        acc += partials[k];
    const float total = block_reduce_sum(acc);
    if (threadIdx.x == 0)
        out[0] = total / (float)n_samples[0];
}

extern "C" void kernel_launch(void* const* d_in, const int* in_sizes, int n_in,
                              void* d_out, int out_size, void* d_ws, size_t ws_size,
                              hipStream_t stream) {
    const float* mu        = (const float*)d_in[0];
    const float* log_sigma = (const float*)d_in[1];
    const float* eps       = (const float*)d_in[2];
    const int*   edges     = (const int*)d_in[3];   // (n_edges, 2) int32
    const int*   n_samp    = (const int*)d_in[4];   // device scalar

    const int n_nodes = in_sizes[0];
    const int n_edges = in_sizes[3] / 2;
    const int n_pairs = n_edges >> 1;

    // Workspace layout: [ c table | block partials ]
    float4* c = (float4*)d_ws;
    size_t c_bytes = (size_t)n_nodes * N_SAMPLES * sizeof(float);
    size_t poff = (c_bytes + 255) & ~(size_t)255;
    float* partials = (float*)((char*)d_ws + poff);

    const int b1 = (n_nodes + TPB - 1) / TPB;
    build_c_kernel<<<b1, TPB, 0, stream>>>(mu, log_sigma, eps, c, n_nodes);

    int b2 = (n_pairs + TPB - 1) / TPB;
    if (b2 > MAX_EDGE_BLOCKS) b2 = MAX_EDGE_BLOCKS;
    if (b2 < 1) b2 = 1;
    edge_reduce_kernel<<<b2, TPB, 0, stream>>>(
        (const v4i*)edges, (const int2*)edges, (const float4*)c,
        partials, n_pairs, n_edges);

    final_reduce_kernel<<<1, TPB, 0, stream>>>(partials, b2, n_samp, (float*)d_out);
}